// BertSelfAttentionWithRelationsRAT_65446711657069
// MI455X (gfx1250) — compile-verified
//
#include <hip/hip_runtime.h>
#include <hip/hip_bf16.h>
#include <math.h>

// ---------------------------------------------------------------------------
// BERT self-attention with relations (RAT), fp32, MI455X (gfx1250, wave32).
// B=4, H=12, S=1024, D=64, Hs=768.
//
//   * 3x GEMM projections with V_WMMA_F32_16X16X4_F32; 32x64 tile per wave
//     (two A sub-tiles share every B fragment -> half the W traffic/FLOP)
//   * fused flash-attention per (b, h, 16-row i-tile):
//       - qrel[i][r] = q_i . rel_k_emb[r]  (7 buckets) precomputed -> LDS
//       - scores tile = WMMA(Q,K^T) + gather(qrel, relation), scaled + mask
//       - online softmax; alpha-rescale work skipped via wave ballot when the
//         running row max did not change (common case)
//       - relacc[i][r] += p  (LDS ds_add_f32 atomics)
//       - ctx += WMMA(P, V); final ctx += relacc @ rel_v_emb ; /= rowsum
// ---------------------------------------------------------------------------

typedef __attribute__((ext_vector_type(2))) float v2f;
typedef __attribute__((ext_vector_type(8))) float v8f;

#define NB 4
#define NH 12
#define NS 1024
#define ND 64
#define NHS 768

static __device__ __forceinline__ v8f wmma_f32(v2f a, v2f b, v8f c) {
  // D(16x16,f32) = A(16x4,f32) x B(4x16,f32) + C
  return __builtin_amdgcn_wmma_f32_16x16x4_f32(false, a, false, b, (short)0, c,
                                               false, false);
}

static __device__ __forceinline__ float rmax16(float v) {
  v = fmaxf(v, __shfl_xor(v, 8));
  v = fmaxf(v, __shfl_xor(v, 4));
  v = fmaxf(v, __shfl_xor(v, 2));
  v = fmaxf(v, __shfl_xor(v, 1));
  return v;  // max over 16-lane half-group (xor<16 never crosses halves)
}
static __device__ __forceinline__ float rsum16(float v) {
  v += __shfl_xor(v, 8);
  v += __shfl_xor(v, 4);
  v += __shfl_xor(v, 2);
  v += __shfl_xor(v, 1);
  return v;
}

// ---------------------------------------------------------------------------
// Kernel 1: QKV projection.  C[4096,768] = X[4096,768] @ W[768,768] + bias,
// stored as [B,H,S,D].  One wave computes a 32x64 output tile: two 16-row
// A sub-tiles reuse every B (weight) fragment.
// grid = (4096/32, 768/64, 3), block = 32.
// ---------------------------------------------------------------------------
__global__ __launch_bounds__(32) void qkv_proj_kernel(
    const float* __restrict__ X, const float* __restrict__ Wq,
    const float* __restrict__ bq, const float* __restrict__ Wk,
    const float* __restrict__ bk, const float* __restrict__ Wv,
    const float* __restrict__ bv, float* __restrict__ qo,
    float* __restrict__ ko, float* __restrict__ vo) {
  const int lane = threadIdx.x;
  const int col = lane & 15;
  const int half = lane >> 4;
  const int m0 = blockIdx.x * 32;
  const int n0 = blockIdx.y * 64;
  const int sel = blockIdx.z;

  const float* W = (sel == 0) ? Wq : ((sel == 1) ? Wk : Wv);
  const float* bias = (sel == 0) ? bq : ((sel == 1) ? bk : bv);
  float* out = (sel == 0) ? qo : ((sel == 1) ? ko : vo);

  const float* xrow0 = X + (size_t)(m0 + col) * NHS;       // A tile 0 row
  const float* xrow1 = X + (size_t)(m0 + 16 + col) * NHS;  // A tile 1 row

  v8f acc[2][4] = {{v8f{}, v8f{}, v8f{}, v8f{}}, {v8f{}, v8f{}, v8f{}, v8f{}}};

#pragma unroll 2
  for (int kb = 0; kb < NHS; kb += 4) {
    // A frags (16x4): lane holds X[row=col][kb + 2*half + {0,1}]
    v2f a0 = *(const v2f*)(xrow0 + kb + 2 * half);
    v2f a1 = *(const v2f*)(xrow1 + kb + 2 * half);
    // B frag (4x16): VGPR0 rows {K0,K2}, VGPR1 rows {K1,K3}
    const float* w0 = W + (size_t)(kb + 2 * half) * NHS + n0 + col;
    const float* w1 = w0 + NHS;
#pragma unroll
    for (int n = 0; n < 4; ++n) {
      v2f bf = {w0[16 * n], w1[16 * n]};
      acc[0][n] = wmma_f32(a0, bf, acc[0][n]);
      acc[1][n] = wmma_f32(a1, bf, acc[1][n]);
    }
  }

  // Epilogue: bias + store in [B,H,S,D] layout.
#pragma unroll
  for (int t = 0; t < 2; ++t) {
#pragma unroll
    for (int n = 0; n < 4; ++n) {
#pragma unroll
      for (int r = 0; r < 8; ++r) {
        const int grow = m0 + 16 * t + r + 8 * half;  // [0,4096): b*1024 + s
        const int b = grow >> 10;
        const int s = grow & 1023;
        const int nn = n0 + 16 * n + col;  // [0,768): h*64 + d
        const int h = nn >> 6;
        const int d = nn & 63;
        out[(((size_t)(b * NH + h) << 10) + s) * ND + d] =
            acc[t][n][r] + bias[nn];
      }
    }
  }
}

// ---------------------------------------------------------------------------
// Kernel 2: fused relation-aware flash attention.
// One wave handles one (b, h, 16-row i-tile); loops over 64 j-tiles of 16.
// grid = (S/16, H, B), block = 32.
// ---------------------------------------------------------------------------
__global__ __launch_bounds__(32) void rat_attn_kernel(
    const float* __restrict__ qb, const float* __restrict__ kb,
    const float* __restrict__ vb, const float* __restrict__ mask,
    const int* __restrict__ relation, const float* __restrict__ relk,
    const float* __restrict__ relv, float* __restrict__ out) {
  const int lane = threadIdx.x;
  const int col = lane & 15;
  const int half = lane >> 4;
  const int i0 = blockIdx.x * 16;
  const int h = blockIdx.y;
  const int b = blockIdx.z;

  __shared__ float s_relk[7][64];
  __shared__ float s_relv[7][64];
  __shared__ float s_qrel[16][8];
  __shared__ float s_relacc[16][8];
  __shared__ float s_alpha[16];
  __shared__ float s_ptile[16][16];

  const size_t head_off = (size_t)(b * NH + h) * NS * ND;
  const float* qh = qb + head_off;
  const float* kh = kb + head_off;
  const float* vh = vb + head_off;

  // Stage the tiny relation tables in LDS; zero relacc.
  for (int idx = lane; idx < 7 * 64; idx += 32) {
    (&s_relk[0][0])[idx] = relk[idx];
    (&s_relv[0][0])[idx] = relv[idx];
  }
  for (int idx = lane; idx < 16 * 8; idx += 32) (&s_relacc[0][0])[idx] = 0.0f;
  __syncthreads();

  // qrel[i][r] = q_i . rel_k_emb[r]
  for (int idx = lane; idx < 16 * 7; idx += 32) {
    const int i = idx / 7;
    const int r = idx % 7;
    const float* qr = qh + (size_t)(i0 + i) * ND;
    float sum = 0.0f;
#pragma unroll 8
    for (int d = 0; d < ND; ++d) sum += qr[d] * s_relk[r][d];
    s_qrel[i][r] = sum;
  }
  if (lane < 16) s_qrel[lane][7] = 0.0f;  // pad bucket (never indexed)
  __syncthreads();

  // Q fragments: A layout, lane holds Q[col][4c + 2*half + {0,1}]
  const float* qrow = qh + (size_t)(i0 + col) * ND;
  v2f qf[16];
#pragma unroll
  for (int c = 0; c < 16; ++c) qf[c] = *(const v2f*)(qrow + 4 * c + 2 * half);

  v8f acc[4] = {v8f{}, v8f{}, v8f{}, v8f{}};
  float mrow[8], lrow[8];
#pragma unroll
  for (int r = 0; r < 8; ++r) {
    mrow[r] = -3.0e38f;
    lrow[r] = 0.0f;
  }
  const float scale = 0.125f;  // 1/sqrt(64)
  const int* relbase = relation + (size_t)b * NS * NS + (size_t)i0 * NS;

  for (int jt = 0; jt < NS / 16; ++jt) {
    const int j0 = jt * 16;

    // Prefetch next tile's K and V rows into cache while we compute.
    if (jt + 1 < NS / 16) {
      __builtin_prefetch(kh + (size_t)(j0 + 16 + col) * ND + 2 * half, 0, 3);
      __builtin_prefetch(vh + (size_t)(j0 + 16 + col) * ND + 2 * half, 0, 3);
    }

    // ---- scores tile: S = Q @ K^T  (16 WMMAs, K=64) ----
    v8f sc = v8f{};
    const float* krow = kh + (size_t)(j0 + col) * ND;
#pragma unroll
    for (int c = 0; c < 16; ++c) {
      v2f kf = *(const v2f*)(krow + 4 * c + 2 * half);
      sc = wmma_f32(qf[c], kf, sc);
    }

    const float mval = mask[b * NS + j0 + col];
    int relidx[8];
    float sm[8];
#pragma unroll
    for (int r = 0; r < 8; ++r) {
      const int row = r + 8 * half;
      const int ri = relbase[(size_t)row * NS + j0 + col];
      relidx[r] = ri;
      sm[r] = (sc[r] + s_qrel[row][ri]) * scale + mval;
    }

    // ---- online softmax update ----
    float alpha[8], p[8];
    bool any_upd = false;
#pragma unroll
    for (int r = 0; r < 8; ++r) {
      const float mt = rmax16(sm[r]);
      const bool upd = mt > mrow[r];
      any_upd |= upd;
      const float mn = upd ? mt : mrow[r];
      alpha[r] = upd ? __expf(mrow[r] - mn) : 1.0f;
      mrow[r] = mn;
      p[r] = __expf(sm[r] - mn);
      lrow[r] = lrow[r] * alpha[r] + rsum16(p[r]);
    }

    if (__ballot(any_upd) != 0ull) {  // wave-uniform: rescale only if needed
#pragma unroll
      for (int n = 0; n < 4; ++n)
#pragma unroll
        for (int r = 0; r < 8; ++r) acc[n][r] *= alpha[r];
      if (col == 0) {
#pragma unroll
        for (int r = 0; r < 8; ++r) s_alpha[r + 8 * half] = alpha[r];
      }
      __syncthreads();
      for (int idx = lane; idx < 16 * 8; idx += 32)
        (&s_relacc[0][0])[idx] *= s_alpha[idx >> 3];
    }
    __syncthreads();

#pragma unroll
    for (int r = 0; r < 8; ++r) {
      const int row = r + 8 * half;
      atomicAdd(&s_relacc[row][relidx[r]], p[r]);  // ds_add_f32
      s_ptile[row][col] = p[r];                    // stage P for transpose
    }
    __syncthreads();

    // ---- ctx += P @ V  (16 WMMAs, K=16, N=64) ----
    v2f pa[4];
#pragma unroll
    for (int c = 0; c < 4; ++c)
      pa[c] = *(const v2f*)(&s_ptile[col][4 * c + 2 * half]);
    const float* vbase = vh + (size_t)j0 * ND;
#pragma unroll
    for (int c = 0; c < 4; ++c) {
      const int vr = 4 * c + 2 * half;
      const float* v0 = vbase + (size_t)vr * ND + col;
      const float* v1 = v0 + ND;
#pragma unroll
      for (int n = 0; n < 4; ++n) {
        v2f vf = {v0[16 * n], v1[16 * n]};
        acc[n] = wmma_f32(pa[c], vf, acc[n]);
      }
    }
  }

  __syncthreads();  // relacc adds complete before epilogue reads

  // ---- epilogue: ctx = (acc + relacc @ rel_v_emb) / rowsum ----
#pragma unroll
  for (int n = 0; n < 4; ++n) {
#pragma unroll
    for (int r = 0; r < 8; ++r) {
      const int row = r + 8 * half;
      const int d = 16 * n + col;
      float cr = 0.0f;
#pragma unroll
      for (int rr = 0; rr < 7; ++rr) cr += s_relacc[row][rr] * s_relv[rr][d];
      const float o = (acc[n][r] + cr) / lrow[r];
      out[(size_t)(b * NS + i0 + row) * NHS + h * ND + d] = o;
    }
  }
}

// ---------------------------------------------------------------------------
extern "C" void kernel_launch(void* const* d_in, const int* in_sizes, int n_in,
                              void* d_out, int out_size, void* d_ws,
                              size_t ws_size, hipStream_t stream) {
  (void)in_sizes; (void)n_in; (void)out_size; (void)ws_size;
  const float* X = (const float*)d_in[0];        // hidden_states [4,1024,768]
  const float* mask = (const float*)d_in[1];     // [4,1,1,1024]
  const int* relation = (const int*)d_in[2];     // [4,1024,1024]
  const float* Wq = (const float*)d_in[3];
  const float* bq = (const float*)d_in[4];
  const float* Wk = (const float*)d_in[5];
  const float* bk = (const float*)d_in[6];
  const float* Wv = (const float*)d_in[7];
  const float* bv = (const float*)d_in[8];
  const float* relk = (const float*)d_in[9];     // [7,64]
  const float* relv = (const float*)d_in[10];    // [7,64]
  float* out = (float*)d_out;

  const size_t per_buf = (size_t)NB * NH * NS * ND;  // 3,145,728 floats (12MB)
  float* q = (float*)d_ws;
  float* k = q + per_buf;
  float* v = k + per_buf;

  qkv_proj_kernel<<<dim3((NB * NS) / 32, NHS / 64, 3), 32, 0, stream>>>(
      X, Wq, bq, Wk, bk, Wv, bv, q, k, v);
  rat_attn_kernel<<<dim3(NS / 16, NH, NB), 32, 0, stream>>>(
      q, k, v, mask, relation, relk, relv, out);
}